// MaxPoolAggregator_58514634441195
// MI455X (gfx1250) — compile-verified
//
#include <hip/hip_runtime.h>

// MaxPool neighbor aggregation: out[q, :] = max_{s<S} features[nbrs[q,s], :]
// N_QUERY=100000, S=10, D_FEAT=128, fp32.
// Bandwidth-bound: 512MB gathered reads + 51MB streaming writes -> ~24us
// floor at 23.3 TB/s. WMMA/TDM do not help (max is not a contraction, and
// TDM-gather would add descriptor setup + an LDS round trip vs. direct
// coalesced global_load_b128 into the consuming VGPRs).
//
// Mapping: one wave32 per 2 queries. Lane L owns floats [4L,4L+4) of a row
// (one b128 per lane per row => each gathered row is a single coalesced 512B
// wave transaction). Indices/row-bases are wave-uniform: readfirstlane makes
// the query id provably uniform so index fetches can go scalar (s_load /
// KMcnt) and row-base math goes SALU; the vector side keeps only the lane*16
// offset. All 2*S row gathers are issued before any max, giving 10KB in
// flight per wave. Output stores are non-temporal so 51MB of streaming writes
// don't evict the reusable feature working set (~323MB unique rows, partial
// reuse in the 192MB L2).

typedef float v4f __attribute__((ext_vector_type(4)));

__device__ __forceinline__ v4f max4(v4f a, v4f b) {
    v4f r;
    r.x = __builtin_fmaxf(a.x, b.x);
    r.y = __builtin_fmaxf(a.y, b.y);
    r.z = __builtin_fmaxf(a.z, b.z);
    r.w = __builtin_fmaxf(a.w, b.w);
    return r;
}

template <int S>
__global__ __launch_bounds__(256) void maxpool_gather_x2(
    const float* __restrict__ features,
    const int* __restrict__ nbrs,
    float* __restrict__ out,
    int n_query)
{
    const int lane = threadIdx.x & 31;
    // Wave id made provably wave-uniform -> scalar index loads + SALU bases.
    const int wave = __builtin_amdgcn_readfirstlane(threadIdx.x >> 5);
    const int pair = blockIdx.x * 8 + wave;

    const int q0 = pair * 2;
    if (q0 >= n_query) return;
    const int q1 = q0 + 1;
    // Clamp so the edge wave computes q0 twice instead of branching divergently.
    const int q1c = (q1 < n_query) ? q1 : q0;

    // Uniform-address index loads (expect s_load via noclobber/uniformity).
    int idx0[S], idx1[S];
#pragma unroll
    for (int s = 0; s < S; ++s) idx0[s] = nbrs[q0 * S + s];
#pragma unroll
    for (int s = 0; s < S; ++s) idx1[s] = nbrs[q1c * S + s];

    // Issue all 2*S gathers back-to-back: 2*S independent global_load_b128,
    // saddr = scalar row base, voffset = lane*16.
    v4f v0[S], v1[S];
#pragma unroll
    for (int s = 0; s < S; ++s) {
        const v4f* row =
            reinterpret_cast<const v4f*>(features + ((long long)idx0[s] << 7));
        v0[s] = row[lane];
    }
#pragma unroll
    for (int s = 0; s < S; ++s) {
        const v4f* row =
            reinterpret_cast<const v4f*>(features + ((long long)idx1[s] << 7));
        v1[s] = row[lane];
    }

    v4f m0 = v0[0];
#pragma unroll
    for (int s = 1; s < S; ++s) m0 = max4(m0, v0[s]);
    v4f m1 = v1[0];
#pragma unroll
    for (int s = 1; s < S; ++s) m1 = max4(m1, v1[s]);

    v4f* orow0 = reinterpret_cast<v4f*>(out + ((long long)q0 << 7));
    __builtin_nontemporal_store(m0, orow0 + lane);
    if (q1 < n_query) {  // uniform branch
        v4f* orow1 = reinterpret_cast<v4f*>(out + ((long long)q1 << 7));
        __builtin_nontemporal_store(m1, orow1 + lane);
    }
}

// Fallback for arbitrary S (running max, one wave per query).
__global__ __launch_bounds__(256) void maxpool_gather_dyn(
    const float* __restrict__ features,
    const int* __restrict__ nbrs,
    float* __restrict__ out,
    int n_query, int S)
{
    const int lane = threadIdx.x & 31;
    const int wave = __builtin_amdgcn_readfirstlane(threadIdx.x >> 5);
    const int q = blockIdx.x * 8 + wave;
    if (q >= n_query) return;

    int i0 = nbrs[q * S];
    const v4f* row0 =
        reinterpret_cast<const v4f*>(features + ((long long)i0 << 7));
    v4f m = row0[lane];
    for (int s = 1; s < S; ++s) {
        int is = nbrs[q * S + s];
        const v4f* row =
            reinterpret_cast<const v4f*>(features + ((long long)is << 7));
        m = max4(m, row[lane]);
    }
    v4f* orow = reinterpret_cast<v4f*>(out + ((long long)q << 7));
    __builtin_nontemporal_store(m, orow + lane);
}

extern "C" void kernel_launch(void* const* d_in, const int* in_sizes, int n_in,
                              void* d_out, int out_size, void* d_ws, size_t ws_size,
                              hipStream_t stream) {
    const float* features = (const float*)d_in[0];  // [N_NODES, 128] fp32
    const int*   nbrs     = (const int*)d_in[1];    // [N_QUERY, S] int32
    float*       out      = (float*)d_out;          // [N_QUERY, 128] fp32

    const int d_feat = 128;                          // fixed by reference
    const int n_query = out_size / d_feat;           // 100000
    const int S = (n_query > 0) ? (in_sizes[1] / n_query) : 0;  // 10
    if (n_query <= 0 || S <= 0) return;

    if (S == 10) {
        const int pairs = (n_query + 1) / 2;         // 2 queries per wave
        const int blocks = (pairs + 7) / 8;          // 8 waves per block
        maxpool_gather_x2<10><<<blocks, 256, 0, stream>>>(
            features, nbrs, out, n_query);
    } else {
        const int blocks = (n_query + 7) / 8;
        maxpool_gather_dyn<<<blocks, 256, 0, stream>>>(
            features, nbrs, out, n_query, S);
    }
}